// GraphConvolution_37907381354545
// MI455X (gfx1250) — compile-verified
//
#include <hip/hip_runtime.h>

typedef float v2f __attribute__((ext_vector_type(2)));
typedef float v4f __attribute__((ext_vector_type(4)));
typedef float v8f __attribute__((ext_vector_type(8)));

static constexpr int kN = 4096;              // adjacency dim
static constexpr int kF = 64;                // feature dim
static constexpr int kWaves = 8;             // 256 threads / block (wave32)
static constexpr int kChunkK = 64;           // K rows staged per chunk
static constexpr int kStride = kChunkK + 4;  // 68 dwords; 68 % 64 == 4 => conflict-minimal b128 reads
static constexpr int kChunks = kN / kChunkK; // 64
static constexpr int kBufDw  = kF * kStride; // 4352 dwords = 17408 B / buffer (x2 = 34816 B LDS)

// One wave computes a 16x64 output tile: O[b, m0:m0+16, :] = A[b,m0:m0+16,:] @ X[b] + bias.
// WMMA f32 16x16x4 fragments (lane L: mr=L%16, h=L/16; fragment-K = 2h+v):
//   K-permutation: one b128 A load at column k+4h feeds two WMMA steps
//     step0 frag-K{2h+v} <- col k+4h+v,  step1 frag-K{2h+v} <- col k+4h+2+v
//   B is staged in LDS transposed [f][k] so the matching 4 K-values per lane are one ds_load_b128.
__global__ __launch_bounds__(256, 1) void gconv_wmma_lds(
    const float* __restrict__ A,     // [8,4096,4096]
    const float* __restrict__ X,     // [8,4096,64]
    const float* __restrict__ bias,  // [64]
    float* __restrict__ O)           // [8,4096,64]
{
  __shared__ __align__(16) float lds[2 * kBufDw];

  const int t    = threadIdx.x;
  const int lane = t & 31;
  const int wave = t >> 5;
  const int tile = blockIdx.x * kWaves + wave;   // 0..2047
  const int b    = tile >> 8;                    // same for all waves in a block
  const int m0   = (tile & 255) << 4;
  const int mr   = lane & 15;
  const int h    = lane >> 4;

  const float* Ap = A + ((size_t)b * kN + (m0 + mr)) * kN + 4 * h;
  const float* Xb = X + (size_t)b * kN * kF;

  // staging coords: thread loads X[kc+kk][fq..fq+3] (rows of 256B, fully coalesced)
  const int kk_s = t >> 4;         // 0..15
  const int fq   = (t & 15) * 4;   // 0..60

  v8f acc0[4] = {};   // 4 N-tiles, step-0 chain
  v8f acc1[4] = {};   // second bank for ILP

  // ---- prologue: stage chunk 0 into buffer 0 (transposed [f][k], stride 68) ----
  {
    v4f cur[4];
#pragma unroll
    for (int p = 0; p < 4; ++p)
      cur[p] = *(const v4f*)(Xb + (size_t)(kk_s + 16 * p) * kF + fq);
#pragma unroll
    for (int p = 0; p < 4; ++p) {
      const int kk = kk_s + 16 * p;
#pragma unroll
      for (int j = 0; j < 4; ++j)
        lds[(fq + j) * kStride + kk] = cur[p][j];
    }
  }
  __syncthreads();

  // ---- main loop: double-buffered chunks ----
  for (int c = 0; c < kChunks; ++c) {
    // issue next chunk's global loads early (overlap with compute)
    v4f nxt[4];
    if (c + 1 < kChunks) {
      const float* Xc = Xb + (size_t)(c + 1) * kChunkK * kF;
#pragma unroll
      for (int p = 0; p < 4; ++p)
        nxt[p] = *(const v4f*)(Xc + (size_t)(kk_s + 16 * p) * kF + fq);
    }

    // compute on current buffer: 8 iterations x (1 global b128 + 4 ds b128 + 8 wmma)
    const float* lp = lds + (c & 1) * kBufDw + mr * kStride + 4 * h;
#pragma unroll
    for (int kk = 0; kk < kChunkK; kk += 8) {
      v4f a = *(const v4f*)(Ap + kk);
      v2f a0 = { a.x, a.y };     // cols kc+kk+4h+{0,1}
      v2f a1 = { a.z, a.w };     // cols kc+kk+4h+{2,3}
#pragma unroll
      for (int f = 0; f < 4; ++f) {
        v4f bb = *(const v4f*)(lp + f * 16 * kStride + kk);  // X rows kc+kk+4h+{0..3}, col f*16+mr
        v2f b0 = { bb.x, bb.y };
        v2f b1 = { bb.z, bb.w };
        acc0[f] = __builtin_amdgcn_wmma_f32_16x16x4_f32(
            false, a0, false, b0, (short)0, acc0[f], false, false);
        acc1[f] = __builtin_amdgcn_wmma_f32_16x16x4_f32(
            false, a1, false, b1, (short)0, acc1[f], false, false);
      }
    }
    Ap += kChunkK;

    // write next chunk into the other buffer
    if (c + 1 < kChunks) {
      float* sp = lds + ((c + 1) & 1) * kBufDw;
#pragma unroll
      for (int p = 0; p < 4; ++p) {
        const int kk = kk_s + 16 * p;
#pragma unroll
        for (int j = 0; j < 4; ++j)
          sp[(fq + j) * kStride + kk] = nxt[p][j];
      }
    }
    __syncthreads();
  }

  // ---- epilogue: C/D VGPR r holds row m0 + 8h + r, column f*16 + mr ----
  float* Orow = O + ((size_t)b * kN + m0 + 8 * h) * kF + mr;
#pragma unroll
  for (int f = 0; f < 4; ++f) {
    const float bv = bias[f * 16 + mr];
#pragma unroll
    for (int r = 0; r < 8; ++r)
      Orow[(size_t)r * kF + f * 16] = acc0[f][r] + acc1[f][r] + bv;
  }
}

extern "C" void kernel_launch(void* const* d_in, const int* in_sizes, int n_in,
                              void* d_out, int out_size, void* d_ws, size_t ws_size,
                              hipStream_t stream) {
  (void)in_sizes; (void)n_in; (void)out_size; (void)d_ws; (void)ws_size;
  const float* A    = (const float*)d_in[0];   // adjacent    [8,4096,4096]
  const float* X    = (const float*)d_in[1];   // annotations [8,4096,64]
  const float* bias = (const float*)d_in[2];   // bias [64]
  float* O = (float*)d_out;                    // [8,4096,64]

  const int totalTiles = 8 * (kN / 16);        // 2048 wave-tiles
  dim3 grid(totalTiles / kWaves);              // 256 blocks
  dim3 block(kWaves * 32);                     // 256 threads
  hipLaunchKernelGGL(gconv_wmma_lds, grid, block, 0, stream, A, X, bias, O);
}